// VGAE_44220983280296
// MI455X (gfx1250) — compile-verified
//
#include <hip/hip_runtime.h>
#include <hip/hip_bf16.h>

typedef __attribute__((ext_vector_type(2))) float v2f;
typedef __attribute__((ext_vector_type(8))) float v8f;

// ---------------- degree / normalization ----------------
__global__ void deg_init_k(float* deg, int n) {
    int i = blockIdx.x * blockDim.x + threadIdx.x;
    if (i < n) deg[i] = 1.0f;                      // self-loop contributes 1
}
__global__ void deg_accum_k(float* deg, const int* __restrict__ dst, int nE) {
    int e = blockIdx.x * blockDim.x + threadIdx.x;
    if (e < nE) atomicAdd(&deg[dst[e]], 1.0f);
}
__global__ void deg_rsqrt_k(float* deg, int n) {
    int i = blockIdx.x * blockDim.x + threadIdx.x;
    if (i < n) deg[i] = rsqrtf(deg[i]);            // deg >= 1 always
}

// ---------------- fp32 WMMA GEMM, LDS-staged B, dinv row-scale epilogue ------
// A: M x K row-major, B: K x N row-major.
// Block = 64 threads (2 waves) -> 32 rows x 64 cols; each wave owns 32x32
// (4 WMMA accumulators). B slice (K x 64) staged in LDS once per block.
// Writes hs = dinv ⊙ (A@B) and agg = hs (self-loop seed).
// grid = (M/32, N/64); requires 32|M, 4|K, 64|N. EXEC all-ones.
__global__ void gemm_scale_wmma(const float* __restrict__ A,
                                const float* __restrict__ B,
                                const float* __restrict__ dinv,
                                float* __restrict__ hs,
                                float* __restrict__ agg,
                                int N, int K) {
    extern __shared__ float Bs[];                  // K x 64
    const int tid  = threadIdx.x;                  // 0..63
    const int lane = tid & 31;
    const int wid  = tid >> 5;                     // 0..1
    const int half = lane >> 4;                    // K-pair select
    const int l15  = lane & 15;
    const int cbase = blockIdx.y * 64;             // first column of block tile

    const size_t row0 = (size_t)blockIdx.x * 32 + l15;   // M sub-tile 0
    const float* __restrict__ a0p = A + row0 * K;
    const float* __restrict__ a1p = a0p + (size_t)16 * K; // M sub-tile 1

    // prefetch the A stream while we fill LDS (global_prefetch_b8)
    __builtin_prefetch(a0p, 0, 0);
    __builtin_prefetch(a1p, 0, 0);

    // cooperative stage of B tile into LDS, float4-coalesced
    const int nq = K * 16;                         // float4 count in tile
    for (int idx = tid; idx < nq; idx += 64) {
        int r = idx >> 4, q = (idx & 15) << 2;
        *(float4*)(Bs + r * 64 + q) =
            *(const float4*)(B + (size_t)r * N + cbase + q);
    }
    __syncthreads();

    // wave's two 16-wide column subtiles within the LDS tile
    const int wc0 = wid * 32 + l15;
    const float* __restrict__ bs0 = Bs + wc0;
    const float* __restrict__ bs1 = Bs + wc0 + 16;

    v8f c00 = {}, c01 = {}, c10 = {}, c11 = {};
    #pragma unroll 4
    for (int k0 = 0; k0 < K; k0 += 4) {
        const int kk = k0 + half * 2;
        v2f a0 = *(const v2f*)(a0p + kk);          // A[m][kk], A[m][kk+1]
        v2f a1 = *(const v2f*)(a1p + kk);
        v2f b0, b1;
        b0.x = bs0[kk * 64];  b0.y = bs0[(kk + 1) * 64];
        b1.x = bs1[kk * 64];  b1.y = bs1[(kk + 1) * 64];
        c00 = __builtin_amdgcn_wmma_f32_16x16x4_f32(false, a0, false, b0,
                                                    (short)0, c00, false, false);
        c01 = __builtin_amdgcn_wmma_f32_16x16x4_f32(false, a0, false, b1,
                                                    (short)0, c01, false, false);
        c10 = __builtin_amdgcn_wmma_f32_16x16x4_f32(false, a1, false, b0,
                                                    (short)0, c10, false, false);
        c11 = __builtin_amdgcn_wmma_f32_16x16x4_f32(false, a1, false, b1,
                                                    (short)0, c11, false, false);
    }

    // C/D layout: VGPR r -> M = half*8 + r, N = lane&15
    const int mb   = blockIdx.x * 32 + half * 8;
    const int col0 = cbase + wid * 32 + l15;
    #pragma unroll
    for (int r = 0; r < 8; ++r) {
        int m = mb + r;
        float dv = dinv[m];
        size_t idx = (size_t)m * N + col0;
        float v0 = c00[r] * dv, v1 = c01[r] * dv;
        hs[idx] = v0;      agg[idx] = v0;
        hs[idx + 16] = v1; agg[idx + 16] = v1;
    }
    #pragma unroll
    for (int r = 0; r < 8; ++r) {
        int m = mb + 16 + r;
        float dv = dinv[m];
        size_t idx = (size_t)m * N + col0;
        float v0 = c10[r] * dv, v1 = c11[r] * dv;
        hs[idx] = v0;      agg[idx] = v0;
        hs[idx + 16] = v1; agg[idx + 16] = v1;
    }
}

// ---------------- edge scatter-add: agg[dst] += hs[src] ----------------
// tpe = F/4 threads per edge; each thread moves a float4, 4 f32 atomics.
__global__ void propagate_k(const float* __restrict__ hs, float* __restrict__ agg,
                            const int* __restrict__ src, const int* __restrict__ dst,
                            int nE, int F, int logTpe) {
    long long gid = (long long)blockIdx.x * blockDim.x + threadIdx.x;
    int e = (int)(gid >> logTpe);
    if (e >= nE) return;
    int t = ((int)gid & ((1 << logTpe) - 1)) << 2;   // feature offset
    int s = src[e], d = dst[e];
    const float4 v = *(const float4*)(hs + (size_t)s * F + t);
    float* p = agg + (size_t)d * F + t;
    atomicAdd(p + 0, v.x);
    atomicAdd(p + 1, v.y);
    atomicAdd(p + 2, v.z);
    atomicAdd(p + 3, v.w);
}

// ---------------- epilogues ----------------
__global__ void finalize_relu_k(const float* __restrict__ agg, const float* __restrict__ dinv,
                                const float* __restrict__ b, float* __restrict__ out,
                                int n, int logF) {
    int i = blockIdx.x * blockDim.x + threadIdx.x;
    if (i >= n) return;
    int m = i >> logF;
    int f = i & ((1 << logF) - 1);
    float v = agg[i] * dinv[m] + b[f];
    out[i] = v > 0.0f ? v : 0.0f;
}

__global__ void finalize_bias_k(float* __restrict__ io, const float* __restrict__ dinv,
                                const float* __restrict__ b, int n, int logF) {
    int i = blockIdx.x * blockDim.x + threadIdx.x;
    if (i >= n) return;
    int m = i >> logF;
    int f = i & ((1 << logF) - 1);
    io[i] = io[i] * dinv[m] + b[f];
}

__global__ void reparam_k(const float* __restrict__ mu, const float* __restrict__ ls,
                          const float* __restrict__ eps, float* __restrict__ z, int n) {
    int i = blockIdx.x * blockDim.x + threadIdx.x;
    if (i >= n) return;
    z[i] = mu[i] + eps[i] * expf(ls[i]);
}

// ---------------- host orchestration ----------------
extern "C" void kernel_launch(void* const* d_in, const int* in_sizes, int n_in,
                              void* d_out, int out_size, void* d_ws, size_t ws_size,
                              hipStream_t stream) {
    const float* x   = (const float*)d_in[0];
    const int*   ei  = (const int*)d_in[1];
    const float* W1  = (const float*)d_in[2];
    const float* b1  = (const float*)d_in[3];
    const float* Wmu = (const float*)d_in[4];
    const float* bmu = (const float*)d_in[5];
    const float* Wls = (const float*)d_in[6];
    const float* bls = (const float*)d_in[7];
    const float* eps = (const float*)d_in[8];
    float* out = (float*)d_out;

    const int IN = 256, HID = 128, LAT = 64;
    const int M = in_sizes[0] / IN;     // 100000
    const int E = in_sizes[1] / 2;      // 1600000
    const int* src = ei;
    const int* dst = ei + E;

    // workspace layout (floats)
    float* ws = (float*)d_ws;
    size_t off = 0;
    float* dinv = ws + off;  off += ((size_t)M + 1023) & ~(size_t)1023;
    float* hs1  = ws + off;  off += (size_t)M * HID;   // layer-1 hs; reused as hidden
    float* agg1 = ws + off;  off += (size_t)M * HID;
    float* hs2  = ws + off;  off += (size_t)M * LAT;   // reused for mu then logstd

    // output slices: (z, mu, logstd) concatenated
    float* z  = out;
    float* mu = out + (size_t)M * LAT;
    float* ls = out + 2 * (size_t)M * LAT;

    // --- degrees -> dinv ---
    deg_init_k <<<(M + 255) / 256, 256, 0, stream>>>(dinv, M);
    deg_accum_k<<<(E + 255) / 256, 256, 0, stream>>>(dinv, dst, E);
    deg_rsqrt_k<<<(M + 255) / 256, 256, 0, stream>>>(dinv, M);

    // --- layer 1 ---
    gemm_scale_wmma<<<dim3(M / 32, HID / 64), 64, (size_t)IN * 64 * sizeof(float), stream>>>(
        x, W1, dinv, hs1, agg1, HID, IN);
    {
        long long nt = (long long)E * (HID / 4);
        propagate_k<<<(unsigned)((nt + 255) / 256), 256, 0, stream>>>(hs1, agg1, src, dst, E, HID, 5);
    }
    finalize_relu_k<<<(M * HID + 255) / 256, 256, 0, stream>>>(agg1, dinv, b1, hs1, M * HID, 7);
    const float* hidden = hs1;

    // --- mu: aggregation lives directly in d_out's mu slice ---
    gemm_scale_wmma<<<dim3(M / 32, LAT / 64), 64, (size_t)HID * 64 * sizeof(float), stream>>>(
        hidden, Wmu, dinv, hs2, mu, LAT, HID);
    {
        long long nt = (long long)E * (LAT / 4);
        propagate_k<<<(unsigned)((nt + 255) / 256), 256, 0, stream>>>(hs2, mu, src, dst, E, LAT, 4);
    }
    finalize_bias_k<<<(M * LAT + 255) / 256, 256, 0, stream>>>(mu, dinv, bmu, M * LAT, 6);

    // --- logstd: reuse hs2 ---
    gemm_scale_wmma<<<dim3(M / 32, LAT / 64), 64, (size_t)HID * 64 * sizeof(float), stream>>>(
        hidden, Wls, dinv, hs2, ls, LAT, HID);
    {
        long long nt = (long long)E * (LAT / 4);
        propagate_k<<<(unsigned)((nt + 255) / 256), 256, 0, stream>>>(hs2, ls, src, dst, E, LAT, 4);
    }
    finalize_bias_k<<<(M * LAT + 255) / 256, 256, 0, stream>>>(ls, dinv, bls, M * LAT, 6);

    // --- z = mu + eps * exp(logstd) ---
    reparam_k<<<(M * LAT + 255) / 256, 256, 0, stream>>>(mu, ls, eps, z, M * LAT);
}